// PairwiseAttentionLayer_77481210020634
// MI455X (gfx1250) — compile-verified
//
#include <hip/hip_runtime.h>
#include <math.h>

typedef __attribute__((ext_vector_type(2))) float v2f;
typedef __attribute__((ext_vector_type(8))) float v8f;

#define N_TGT 8192
#define DEG   32
#define H     512
#define NH    8
#define HD    64
#define HID2  1024

// ---------------- GEMM with f32 WMMA (16x16x4) ----------------
// C[M,N] = act(A[M,K] @ W[K,N] + bias [+ res]) [* mask(row)]
// Block tile 128x64, BK=32, 8 waves each computing a 32x32 sub-tile
// (4 wmma accumulators). Global->register prefetch is software-pipelined
// so HBM/L2 latency overlaps the 32-WMMA compute phase per K-tile.
#define BM 128
#define BN 64
#define BK 32
#define LDA 36   // padded LDS stride for A tile (128 x 36), 16B-aligned rows
#define LDB 68   // padded LDS stride for B tile (32 x 68), 16B-aligned rows

__global__ __launch_bounds__(256)
void gemm_wmma_f32(const float* __restrict__ A, const float* __restrict__ W,
                   const float* __restrict__ bias, const float* __restrict__ res,
                   const float* __restrict__ mask, float* __restrict__ C,
                   int K, int Ncols, int relu)
{
    __shared__ __align__(16) float As[BM * LDA];   // 4608 floats
    __shared__ __align__(16) float Bs[BK * LDB];   // 2176 floats

    const int t    = threadIdx.x;
    const int row0 = blockIdx.x * BM;
    const int col0 = blockIdx.y * BN;

    const int lane = t & 31;
    const int w    = t >> 5;             // 8 waves
    const int rowT = (w & 3) * 32;       // wave row tile (32 rows)
    const int colT = (w >> 2) * 32;      // wave col tile (32 cols)
    const int lo   = lane & 15;
    const int hi   = lane >> 4;

    v8f acc00 = {0.f, 0.f, 0.f, 0.f, 0.f, 0.f, 0.f, 0.f};
    v8f acc01 = acc00, acc10 = acc00, acc11 = acc00;

    // staging: A tile 128x32 = 4096 floats -> 16 floats (4x float4) / thread
    //          B tile  32x64 = 2048 floats ->  8 floats (2x float4) / thread
    const int ar = t >> 1;               // 0..127
    const int ac = (t & 1) << 4;         // 0 or 16
    const int br = t >> 3;               // 0..31
    const int bc = (t & 7) << 3;         // 0..56

    float4 a_r0, a_r1, a_r2, a_r3, b_r0, b_r1;
    {
        const float* ag = A + (size_t)(row0 + ar) * K + ac;
        a_r0 = *(const float4*)(ag);
        a_r1 = *(const float4*)(ag + 4);
        a_r2 = *(const float4*)(ag + 8);
        a_r3 = *(const float4*)(ag + 12);
        const float* bg = W + (size_t)br * Ncols + col0 + bc;
        b_r0 = *(const float4*)(bg);
        b_r1 = *(const float4*)(bg + 4);
    }

    for (int k0 = 0; k0 < K; k0 += BK) {
        __syncthreads();                 // protect previous iteration's reads
        {
            float* as = &As[ar * LDA + ac];
            *(float4*)(as)      = a_r0;
            *(float4*)(as + 4)  = a_r1;
            *(float4*)(as + 8)  = a_r2;
            *(float4*)(as + 12) = a_r3;
            float* bs = &Bs[br * LDB + bc];
            *(float4*)(bs)     = b_r0;
            *(float4*)(bs + 4) = b_r1;
        }
        __syncthreads();

        // prefetch next K-tile into registers; overlaps with WMMAs below
        const int kn = k0 + BK;
        if (kn < K) {
            const float* ag = A + (size_t)(row0 + ar) * K + kn + ac;
            a_r0 = *(const float4*)(ag);
            a_r1 = *(const float4*)(ag + 4);
            a_r2 = *(const float4*)(ag + 8);
            a_r3 = *(const float4*)(ag + 12);
            const float* bg = W + (size_t)(kn + br) * Ncols + col0 + bc;
            b_r0 = *(const float4*)(bg);
            b_r1 = *(const float4*)(bg + 4);
        }

        // A frag: lane(m) holds K = kc + {0,1} (lanes 0-15) / {2,3} (lanes 16-31)
        const float* ap0 = &As[(rowT + lo) * LDA + 2 * hi];
        const float* ap1 = ap0 + 16 * LDA;
        // B frag: lane(n) holds K rows kc+2*hi and kc+1+2*hi
        const float* bp = &Bs[(2 * hi) * LDB + colT + lo];
        #pragma unroll
        for (int kc = 0; kc < BK; kc += 4) {
            v2f a0 = *(const v2f*)(ap0 + kc);
            v2f a1 = *(const v2f*)(ap1 + kc);
            v2f b0, b1;
            b0.x = bp[kc * LDB];
            b0.y = bp[(kc + 1) * LDB];
            b1.x = bp[kc * LDB + 16];
            b1.y = bp[(kc + 1) * LDB + 16];
            acc00 = __builtin_amdgcn_wmma_f32_16x16x4_f32(false, a0, false, b0,
                                                          (short)0, acc00, false, false);
            acc01 = __builtin_amdgcn_wmma_f32_16x16x4_f32(false, a0, false, b1,
                                                          (short)0, acc01, false, false);
            acc10 = __builtin_amdgcn_wmma_f32_16x16x4_f32(false, a1, false, b0,
                                                          (short)0, acc10, false, false);
            acc11 = __builtin_amdgcn_wmma_f32_16x16x4_f32(false, a1, false, b1,
                                                          (short)0, acc11, false, false);
        }
    }

    // D layout: VGPR r -> row = r + 8*hi (+0/+16), col = lo (+0/+16)
    const int ca = col0 + colT + lo;
    const int cb = ca + 16;
    const float ba = bias[ca];
    const float bb = bias[cb];
    #pragma unroll
    for (int r = 0; r < 8; ++r) {
        const int rowA = row0 + rowT + r + 8 * hi;
        const int rowB = rowA + 16;
        float v00 = acc00[r] + ba;
        float v01 = acc01[r] + bb;
        float v10 = acc10[r] + ba;
        float v11 = acc11[r] + bb;
        if (res) {
            v00 += res[(size_t)rowA * Ncols + ca];
            v01 += res[(size_t)rowA * Ncols + cb];
            v10 += res[(size_t)rowB * Ncols + ca];
            v11 += res[(size_t)rowB * Ncols + cb];
        }
        if (relu) {
            v00 = fmaxf(v00, 0.f);
            v01 = fmaxf(v01, 0.f);
            v10 = fmaxf(v10, 0.f);
            v11 = fmaxf(v11, 0.f);
        }
        if (mask) {
            const float ma = mask[rowA];
            const float mb = mask[rowB];
            v00 *= ma; v01 *= ma;
            v10 *= mb; v11 *= mb;
        }
        C[(size_t)rowA * Ncols + ca] = v00;
        C[(size_t)rowA * Ncols + cb] = v01;
        C[(size_t)rowB * Ncols + ca] = v10;
        C[(size_t)rowB * Ncols + cb] = v11;
    }
}

// ---------------- Edge attention (segment softmax over 32 edges/target) ----------------
__global__ __launch_bounds__(256)
void edge_attn(const float* __restrict__ Q, const float* __restrict__ Kb,
               const float* __restrict__ Vb, const int* __restrict__ src_idx,
               float* __restrict__ Ctx)
{
    const int i = blockIdx.x;
    const int t = threadIdx.x;
    __shared__ __align__(16) float qs[H];
    __shared__ int   srcs[DEG];
    __shared__ float sc[DEG * NH];
    __shared__ float red[DEG * NH];

    qs[t]       = Q[(size_t)i * H + t];
    qs[t + 256] = Q[(size_t)i * H + t + 256];
    if (t < DEG) srcs[t] = src_idx[i * DEG + t];
    __syncthreads();

    // one (edge, head) dot product per thread: t = e*8 + h
    const int h = t & 7;
    {
        const int e = t >> 3;
        const float* kp = Kb + (size_t)srcs[e] * H + h * HD;
        const float* qp = &qs[h * HD];
        float acc = 0.f;
        #pragma unroll
        for (int j = 0; j < HD; j += 4) {
            float4 kk = *(const float4*)(kp + j);
            float4 qq = *(const float4*)(qp + j);
            acc += kk.x * qq.x + kk.y * qq.y + kk.z * qq.z + kk.w * qq.w;
        }
        sc[t]  = acc * 0.125f;   // 1/sqrt(64)
        red[t] = sc[t];
    }
    __syncthreads();

    // segment max per head (reduce over edges; stride multiples of 8 keep head fixed)
    for (int off = 128; off >= 8; off >>= 1) {
        if (t < off) red[t] = fmaxf(red[t], red[t + off]);
        __syncthreads();
    }
    const float mx = red[h];
    __syncthreads();
    const float p = expf(sc[t] - mx);
    red[t] = p;
    __syncthreads();
    for (int off = 128; off >= 8; off >>= 1) {
        if (t < off) red[t] += red[t + off];
        __syncthreads();
    }
    const float denom = red[h];
    sc[t] = p / fmaxf(denom, 1e-9f);
    __syncthreads();

    // context: each thread accumulates 2 channels over 32 edges
    const int c  = t * 2;
    const int hh = c >> 6;
    float ax = 0.f, ay = 0.f;
    #pragma unroll 8
    for (int e2 = 0; e2 < DEG; ++e2) {
        const float a = sc[e2 * NH + hh];
        const float2 vv = *(const float2*)(Vb + (size_t)srcs[e2] * H + c);
        ax += a * vv.x;
        ay += a * vv.y;
    }
    float2 o;
    o.x = ax;
    o.y = ay;
    *(float2*)(Ctx + (size_t)i * H + c) = o;
}

// ---------------- LayerNorm (two-pass, matches reference) ----------------
__global__ __launch_bounds__(256)
void layernorm_k(const float* __restrict__ X, const float* __restrict__ g,
                 const float* __restrict__ b, float* __restrict__ Y)
{
    const int i = blockIdx.x;
    const int t = threadIdx.x;
    __shared__ float red[256];
    const float x0 = X[(size_t)i * H + t];
    const float x1 = X[(size_t)i * H + t + 256];
    red[t] = x0 + x1;
    __syncthreads();
    for (int off = 128; off >= 1; off >>= 1) {
        if (t < off) red[t] += red[t + off];
        __syncthreads();
    }
    const float mu = red[0] * (1.f / H);
    __syncthreads();
    const float d0 = x0 - mu, d1 = x1 - mu;
    red[t] = d0 * d0 + d1 * d1;
    __syncthreads();
    for (int off = 128; off >= 1; off >>= 1) {
        if (t < off) red[t] += red[t + off];
        __syncthreads();
    }
    const float rs = rsqrtf(red[0] * (1.f / H) + 1e-5f);
    Y[(size_t)i * H + t]       = d0 * rs * g[t] + b[t];
    Y[(size_t)i * H + t + 256] = d1 * rs * g[t + 256] + b[t + 256];
}

// ---------------- attending-row mask ----------------
__global__ void zero_f32(float* __restrict__ p, int n)
{
    int i = blockIdx.x * 256 + threadIdx.x;
    if (i < n) p[i] = 0.f;
}
__global__ void scatter_ones(const int* __restrict__ tgt, float* __restrict__ m, int e)
{
    int i = blockIdx.x * 256 + threadIdx.x;
    if (i < e) m[tgt[i]] = 1.f;
}

extern "C" void kernel_launch(void* const* d_in, const int* in_sizes, int n_in,
                              void* d_out, int out_size, void* d_ws, size_t ws_size,
                              hipStream_t stream)
{
    (void)in_sizes; (void)n_in; (void)out_size; (void)ws_size;
    const float* h_target = (const float*)d_in[0];
    const float* h_src_n  = (const float*)d_in[1];
    const float* h_tgt_n  = (const float*)d_in[2];
    const int*   tgt_idx  = (const int*)d_in[3];
    const int*   src_idx  = (const int*)d_in[4];
    const float* Wq = (const float*)d_in[5];  const float* bq = (const float*)d_in[6];
    const float* Wk = (const float*)d_in[7];  const float* bk = (const float*)d_in[8];
    const float* Wv = (const float*)d_in[9];  const float* bv = (const float*)d_in[10];
    const float* Wo = (const float*)d_in[11]; const float* bo = (const float*)d_in[12];
    const float* ln_g = (const float*)d_in[13];
    const float* ln_b = (const float*)d_in[14];
    const float* W1 = (const float*)d_in[15]; const float* b1 = (const float*)d_in[16];
    const float* W2 = (const float*)d_in[17]; const float* b2 = (const float*)d_in[18];

    float* ws = (float*)d_ws;
    const size_t NS = (size_t)N_TGT * H;
    float* qb  = ws;            // N x H
    float* kb  = ws + 1 * NS;   // M x H
    float* vb  = ws + 2 * NS;   // M x H
    float* ctx = ws + 3 * NS;   // N x H
    float* h2  = ws + 4 * NS;   // N x H (attn out + residual)
    float* hn  = ws + 5 * NS;   // N x H (layernorm)
    float* f1  = ws + 6 * NS;   // N x HID2 (2*NS)
    float* mk  = ws + 8 * NS;   // N

    dim3 blk(256);
    dim3 gP(N_TGT / BM, H / BN);           // 64 x 8
    dim3 gF1(N_TGT / BM, HID2 / BN);       // 64 x 16

    // Q/K/V projections
    hipLaunchKernelGGL(gemm_wmma_f32, gP, blk, 0, stream,
                       h_tgt_n, Wq, bq, (const float*)nullptr, (const float*)nullptr, qb, H, H, 0);
    hipLaunchKernelGGL(gemm_wmma_f32, gP, blk, 0, stream,
                       h_src_n, Wk, bk, (const float*)nullptr, (const float*)nullptr, kb, H, H, 0);
    hipLaunchKernelGGL(gemm_wmma_f32, gP, blk, 0, stream,
                       h_src_n, Wv, bv, (const float*)nullptr, (const float*)nullptr, vb, H, H, 0);
    // sparse attention (one block per target)
    hipLaunchKernelGGL(edge_attn, dim3(N_TGT), blk, 0, stream, qb, kb, vb, src_idx, ctx);
    // output projection + residual
    hipLaunchKernelGGL(gemm_wmma_f32, gP, blk, 0, stream,
                       ctx, Wo, bo, h_target, (const float*)nullptr, h2, H, H, 0);
    // layernorm
    hipLaunchKernelGGL(layernorm_k, dim3(N_TGT), blk, 0, stream, h2, ln_g, ln_b, hn);
    // FFN layer 1 with ReLU
    hipLaunchKernelGGL(gemm_wmma_f32, gF1, blk, 0, stream,
                       hn, W1, b1, (const float*)nullptr, (const float*)nullptr, f1, H, HID2, 1);
    // attending mask
    hipLaunchKernelGGL(zero_f32, dim3((N_TGT + 255) / 256), blk, 0, stream, mk, N_TGT);
    hipLaunchKernelGGL(scatter_ones, dim3((N_TGT * DEG + 255) / 256), blk, 0, stream,
                       tgt_idx, mk, N_TGT * DEG);
    // FFN layer 2 + residual + mask -> output
    hipLaunchKernelGGL(gemm_wmma_f32, gP, blk, 0, stream,
                       f1, W2, b2, h2, mk, (float*)d_out, HID2, H, 0);
}